// MyTreeLSTMUnit_78383153152086
// MI455X (gfx1250) — compile-verified
//
#include <hip/hip_runtime.h>
#include <stdint.h>

typedef float v2f __attribute__((ext_vector_type(2)));
typedef float v8f __attribute__((ext_vector_type(8)));

#define B_TOTAL   65536
#define IN_DIM    256
#define M_DIM     256
#define KTOT      1280
#define BT        128                 // batch rows per workgroup
#define KC        128                 // K-chunk staged in LDS
#define LDS_STRIDE 132                // dwords/row: 16B-aligned rows, bank-conflict-free
#define NCHUNK    10
#define BUF_ELEMS (BT * LDS_STRIDE)   // 16,896 floats = 67,584 B per buffer

__device__ __forceinline__ float fast_sigmoid(float x) { return 1.0f / (1.0f + __expf(-x)); }
__device__ __forceinline__ float fast_tanh(float x)    { float e = __expf(2.0f * x); return (e - 1.0f) / (e + 1.0f); }

__device__ __forceinline__ v8f splat8(float x) {
    v8f v;
#pragma unroll
    for (int e = 0; e < 8; ++e) v[e] = x;
    return v;
}

// Weight fragment load: uniform SGPR base + unsigned 32-bit *byte* offset
// -> matches the global_load saddr (GVS) pattern, 24-bit imm for kk.
__device__ __forceinline__ v2f ldg2(const float* base, unsigned byteoff) {
    return *(const v2f*)((const char*)base + byteoff);
}

// --- CDNA5 async global->LDS staging (ASYNCcnt), GVS addressing ---
// INST_OFFSET is added to BOTH the LDS and global addresses (ISA 08 §4.4):
// one LDS-addr VGPR + one 32-bit global-offset VGPR covers 128 contiguous bytes.
__device__ __forceinline__ void stage_async128B(uint32_t lds_byte, const float* src, uint32_t gbyte)
{
    unsigned long long s = (unsigned long long)(uintptr_t)src;
    asm volatile(
        "global_load_async_to_lds_b128 %0, %1, %2\n\t"
        "global_load_async_to_lds_b128 %0, %1, %2 offset:16\n\t"
        "global_load_async_to_lds_b128 %0, %1, %2 offset:32\n\t"
        "global_load_async_to_lds_b128 %0, %1, %2 offset:48\n\t"
        "global_load_async_to_lds_b128 %0, %1, %2 offset:64\n\t"
        "global_load_async_to_lds_b128 %0, %1, %2 offset:80\n\t"
        "global_load_async_to_lds_b128 %0, %1, %2 offset:96\n\t"
        "global_load_async_to_lds_b128 %0, %1, %2 offset:112"
        :: "v"(lds_byte), "v"(gbyte), "s"(s) : "memory");
}

__device__ __forceinline__ void wait_async0() {
    asm volatile("s_wait_asynccnt 0x0" ::: "memory");
}

#define WMMA_F32(accv, av, bv) \
    accv = __builtin_amdgcn_wmma_f32_16x16x4_f32(false, av, false, bv, (short)0, accv, false, false)

struct Acc { v8f t[3][2][4]; };   // [gate][colblock][rowblock] = 24 tiles, 192 VGPRs

// 4 row-block WMMAs sharing one B fragment
#define WMMA_ROW(gc, bv) \
    WMMA_F32(gc[0], a0, bv); WMMA_F32(gc[1], a1, bv); \
    WMMA_F32(gc[2], a2, bv); WMMA_F32(gc[3], a3, bv)

__device__ __forceinline__ void chunk_mma(const float* buf, int lb,
                                          const float* __restrict__ w0,
                                          const float* __restrict__ w1,
                                          const float* __restrict__ w2,
                                          unsigned ob0, unsigned ob1, Acc& A)
{
#pragma unroll 2
    for (int kk = 0; kk < KC; kk += 4) {
        const unsigned kb = (unsigned)(kk * 4);
        // A fragments: one LDS base + 16-bit imm offsets (row spacing 8448 B, kk*4 <= 508 B)
        v2f a0 = *(const v2f*)(buf + lb +                   kk);
        v2f a1 = *(const v2f*)(buf + lb + 16 * LDS_STRIDE + kk);
        v2f a2 = *(const v2f*)(buf + lb + 32 * LDS_STRIDE + kk);
        v2f a3 = *(const v2f*)(buf + lb + 48 * LDS_STRIDE + kk);
        // B fragments: saddr + invariant 32-bit byte voffset (weights L2-resident)
        v2f b00 = ldg2(w0, ob0 + kb);
        v2f b01 = ldg2(w0, ob1 + kb);
        v2f b10 = ldg2(w1, ob0 + kb);
        v2f b11 = ldg2(w1, ob1 + kb);
        v2f b20 = ldg2(w2, ob0 + kb);
        v2f b21 = ldg2(w2, ob1 + kb);
        // 24 WMMAs: each B fragment feeds 4 row-blocks
        WMMA_ROW(A.t[0][0], b00);
        WMMA_ROW(A.t[0][1], b01);
        WMMA_ROW(A.t[1][0], b10);
        WMMA_ROW(A.t[1][1], b11);
        WMMA_ROW(A.t[2][0], b20);
        WMMA_ROW(A.t[2][1], b21);
    }
}

__global__ __launch_bounds__(512, 1)
void treelstm_wmma_kernel(const float* __restrict__ inputs,
                          const float* __restrict__ children,
                          const float* __restrict__ Wi, const float* __restrict__ bi,
                          const float* __restrict__ Wo, const float* __restrict__ bo,
                          const float* __restrict__ Wu, const float* __restrict__ bu,
                          const float* __restrict__ Ui,
                          const float* __restrict__ Uo,
                          const float* __restrict__ Uu,
                          float* __restrict__ out)
{
    __shared__ float lds[2 * BUF_ELEMS];   // double-buffered A chunk, 135,168 B

    const int tid  = threadIdx.x;
    const int lane = tid & 31;               // wave32
    const int wave = tid >> 5;               // 0..15
    const int btbase = blockIdx.x * BT;

    // wave -> (4 row-blocks, 2 col-blocks), all 3 gates concurrently
    const int wgrp = wave & 1;               // row half: rows 0-63 or 64-127
    const int cb0  = (wave >> 1) * 2;        // col blocks {cb0, cb0+1}
    const int ml   = lane & 15;
    const int khi  = (lane >> 4) * 2;        // K sub-offset: 0 or 2

    const int n0 = cb0 * 16 + ml;
    const int n1 = n0 + 16;

    // loop-invariant per-lane offsets
    const int      lb   = (wgrp * 64 + ml) * LDS_STRIDE + khi;  // LDS dword base (rb stride via imm)
    const unsigned nwb0 = (unsigned)((n0 * M_DIM + khi) * 4);   // weight byte offset, col block 0
    const unsigned nwb1 = (unsigned)((n1 * M_DIM + khi) * 4);   // weight byte offset, col block 1

    // accumulators initialized with bias broadcast (lane fixes N)
    Acc A;
    {
        const v8f si0 = splat8(bi[n0]), si1 = splat8(bi[n1]);
        const v8f so0 = splat8(bo[n0]), so1 = splat8(bo[n1]);
        const v8f su0 = splat8(bu[n0]), su1 = splat8(bu[n1]);
#pragma unroll
        for (int r = 0; r < 4; ++r) {
            A.t[0][0][r] = si0; A.t[0][1][r] = si1;
            A.t[1][0][r] = so0; A.t[1][1][r] = so1;
            A.t[2][0][r] = su0; A.t[2][1][r] = su1;
        }
    }

    // --- async A staging: 512 threads x 128 B = 64 KB chunk, no VGPR data path ---
    const int srow = tid >> 2;               // 0..127 (4 threads per row)
    const int scol = (tid & 3) * 32;         // 32 floats (128 B) per thread
    const uint32_t lds_byte0 = (uint32_t)(uintptr_t)&lds[srow * LDS_STRIDE + scol];

    // issue chunk 0 (inputs) into buffer 0
    stage_async128B(lds_byte0, inputs, (uint32_t)(((btbase + srow) * IN_DIM + scol) * 4));

    for (int c = 0; c < NCHUNK; ++c) {
        wait_async0();        // this wave's slice of chunk c landed in LDS
        __syncthreads();      // all waves' slices landed; other buffer free for reuse

        // kick off the async DMA for chunk c+1 into the other buffer
        if (c + 1 < NCHUNK) {
            const int k0n = (c + 1) * KC;
            const uint32_t dstb = lds_byte0 + (uint32_t)(((c + 1) & 1) * BUF_ELEMS * 4);
            if (k0n < IN_DIM) {
                stage_async128B(dstb, inputs,
                                (uint32_t)(((btbase + srow) * IN_DIM + k0n + scol) * 4));
            } else {
                const int kch = (k0n - IN_DIM) >> 8;
                const int m0  = (k0n - IN_DIM) & 255;
                stage_async128B(dstb, children,
                                (uint32_t)(((kch * B_TOTAL + btbase + srow) * (2 * M_DIM) + m0 + scol) * 4));
                // warm L2 for next chunk's weight row for this lane
                __builtin_prefetch((const char*)Ui + (unsigned)((kch * (M_DIM * M_DIM) + m0) * 4) + nwb0, 0, 1);
            }
        }

        // MMA over this chunk: uniform branch selects W-phase vs U-phase
        const float* buf = lds + (c & 1) * BUF_ELEMS;
        if (c < 2) {
            const unsigned wb = (unsigned)(c * KC * 4);
            chunk_mma(buf, lb, Wi, Wo, Wu, wb + nwb0, wb + nwb1, A);
        } else {
            const int kch = (c - 2) >> 1;
            const int m0  = ((c - 2) & 1) * KC;
            const unsigned wb = (unsigned)((kch * (M_DIM * M_DIM) + m0) * 4);
            chunk_mma(buf, lb, Ui, Uo, Uu, wb + nwb0, wb + nwb1, A);
        }
    }

    // --- epilogue: c = sigmoid(i)*tanh(u); h = sigmoid(o)*tanh(c); out = [h | c] ---
#pragma unroll
    for (int cj = 0; cj < 2; ++cj) {
        const int n = cj ? n1 : n0;
#pragma unroll
        for (int r = 0; r < 4; ++r) {
            const int rowbase = btbase + wgrp * 64 + r * 16 + (lane >> 4) * 8;
#pragma unroll
            for (int v = 0; v < 8; ++v) {
                const float iv = A.t[0][cj][r][v];
                const float ov = A.t[1][cj][r][v];
                const float uv = A.t[2][cj][r][v];
                const float cc = fast_sigmoid(iv) * fast_tanh(uv);
                const float hh = fast_sigmoid(ov) * fast_tanh(cc);
                const int row = rowbase + v;
                out[row * (2 * M_DIM) + n]         = hh;
                out[row * (2 * M_DIM) + M_DIM + n] = cc;
            }
        }
    }
}

extern "C" void kernel_launch(void* const* d_in, const int* in_sizes, int n_in,
                              void* d_out, int out_size, void* d_ws, size_t ws_size,
                              hipStream_t stream) {
    const float* inputs   = (const float*)d_in[0];
    const float* children = (const float*)d_in[1];
    // d_in[2] = arities (unused by the reference math)
    const float* Wi = (const float*)d_in[3];
    const float* bi = (const float*)d_in[4];
    const float* Wo = (const float*)d_in[5];
    const float* bo = (const float*)d_in[6];
    const float* Wu = (const float*)d_in[7];
    const float* bu = (const float*)d_in[8];
    // d_in[9] = Wf, d_in[10] = bf (unused)
    const float* Ui = (const float*)d_in[11];
    const float* Uo = (const float*)d_in[12];
    const float* Uu = (const float*)d_in[13];
    // d_in[14] = Uf (unused)
    float* out = (float*)d_out;

    dim3 grid(B_TOTAL / BT);   // 512 workgroups
    dim3 block(512);           // 16 wave32 waves -> 4 waves/SIMD, 256-VGPR budget
    hipLaunchKernelGGL(treelstm_wmma_kernel, grid, block, 0, stream,
                       inputs, children, Wi, bi, Wo, bo, Wu, bu, Ui, Uo, Uu, out);
}